// DecoderRNN_62337155334717
// MI455X (gfx1250) — compile-verified
//
#include <hip/hip_runtime.h>
#include <math.h>

#define B_ 16
#define S_ 100
#define V_ 30000
#define H_ 512
#define E_ 300
#define T_ 50
#define ML_ 50
#define D_ (3*H_+E_)      /* 1836 */
#define G3_ (3*H_)        /* 1536 */
#define VS_ (V_+S_)       /* 30100 */
#define NEGV (-1000000.0f)

typedef float v2f __attribute__((ext_vector_type(2)));
typedef float v8f __attribute__((ext_vector_type(8)));
typedef unsigned long long ull;

// ---------------------------------------------------------------------------
// fp32 WMMA 16x16 tile: D(16 x N) tile n0 = X(16 x K) @ W(N x K)^T + bias
// ---------------------------------------------------------------------------
__device__ __forceinline__ void gemm_tile(const float* __restrict__ X, int ldx,
                                          const float* __restrict__ W, int ldw,
                                          float* __restrict__ Dst, int ldd,
                                          int K, int tile,
                                          const float* __restrict__ bias,
                                          int neg_first, int lane) {
    const int l15  = lane & 15;
    const int half = lane >> 4;                 // 0 or 1
    const int koff = half << 1;                 // lanes 16-31 carry K+2,K+3
    const int n0   = tile * 16;

    const float* xrow = X + l15 * ldx + koff;          // A: row M = l15
    const float* wrow = W + (n0 + l15) * ldw + koff;   // B: col N = n0+l15

    v8f acc = {0.f,0.f,0.f,0.f,0.f,0.f,0.f,0.f};
    #pragma unroll 4
    for (int k = 0; k < K; k += 4) {
        v2f a = { xrow[k], xrow[k+1] };
        v2f b = { wrow[k], wrow[k+1] };
        acc = __builtin_amdgcn_wmma_f32_16x16x4_f32(
                  false, a, false, b, (short)0, acc, false, false);
    }

    const int n = n0 + l15;
    const float bv = bias ? bias[n] : 0.0f;
    #pragma unroll
    for (int v = 0; v < 8; ++v) {
        const int m = v + (half << 3);          // C/D: VGPR v -> row, lane -> col
        float val = acc[v] + bv;
        if (neg_first && n == 0) val = NEGV;
        Dst[m * ldd + n] = val;
    }
}

// Single-problem WMMA GEMM (used for the big vocab projection)
__global__ void k_gemm_wmma(const float* __restrict__ X, int ldx,
                            const float* __restrict__ W, int ldw,
                            float* __restrict__ Dst, int ldd,
                            int K, int ntiles,
                            const float* __restrict__ bias, int neg_first) {
    const int wave = blockIdx.x * (blockDim.x >> 5) + (threadIdx.x >> 5);
    if (wave >= ntiles) return;                 // uniform per wave
    gemm_tile(X, ldx, W, ldw, Dst, ldd, K, wave, bias, neg_first, threadIdx.x & 31);
}

// Two-problem fused WMMA GEMM (wave-uniform problem select)
__global__ void k_gemm_wmma2(const float* __restrict__ X0, int ldx0,
                             const float* __restrict__ W0, int ldw0,
                             float* __restrict__ D0, int ldd0, int K0, int nt0,
                             const float* __restrict__ b0,
                             const float* __restrict__ X1, int ldx1,
                             const float* __restrict__ W1, int ldw1,
                             float* __restrict__ D1, int ldd1, int K1, int nt1,
                             const float* __restrict__ b1) {
    const int wave = blockIdx.x * (blockDim.x >> 5) + (threadIdx.x >> 5);
    const int lane = threadIdx.x & 31;
    if (wave < nt0) {
        gemm_tile(X0, ldx0, W0, ldw0, D0, ldd0, K0, wave, b0, 0, lane);
    } else if (wave < nt0 + nt1) {
        gemm_tile(X1, ldx1, W1, ldw1, D1, ldd1, K1, wave - nt0, b1, 0, lane);
    }
}

// ---------------------------------------------------------------------------
// Init kernels
// ---------------------------------------------------------------------------
__global__ void k_init_big(float* __restrict__ dout, const float* __restrict__ z,
                           float* __restrict__ hidden, float* __restrict__ pre_hidden,
                           float* __restrict__ sel) {
    int tid = blockIdx.x * blockDim.x + threadIdx.x;
    if (tid >= B_ * VS_) return;
    int b = tid / VS_, i = tid - b * VS_;
    dout[(size_t)b * ML_ * VS_ + i] = (i == 1) ? 1.0f : 0.0f;   // SOS row (t=0)
    if (i < H_) {
        hidden[b * H_ + i]     = z[b * H_ + i];
        pre_hidden[b * H_ + i] = z[b * H_ + i];
        sel[b * H_ + i]        = 0.0f;
    }
}

__global__ void k_init_state(int* st, const int* __restrict__ templates,
                             float* __restrict__ out_idxs) {
    int b = threadIdx.x;
    if (b >= B_) return;
    int* sampled = st, *pre_sampled = st + B_, *eos = st + 2*B_, *pre_seg = st + 3*B_;
    int* index = st + 4*B_, *c_len = st + 5*B_;
    sampled[b] = 1; pre_sampled[b] = 1; eos[b] = 1;
    pre_seg[b] = templates[b * T_];
    index[b] = 0; c_len[b] = 0;
    out_idxs[b * ML_] = 1.0f;
}

// ---------------------------------------------------------------------------
// Per-step control state update (scalar per batch row)
// ---------------------------------------------------------------------------
__global__ void k_ctrl(int* st, const int* __restrict__ templates,
                       float* __restrict__ out_idxs, int t) {
    int b = threadIdx.x;
    if (b >= B_) return;
    int* sampled = st, *pre_sampled = st + B_, *eos = st + 2*B_, *pre_seg = st + 3*B_;
    int* index = st + 4*B_, *c_len = st + 5*B_, *keep = st + 6*B_, *embidx = st + 7*B_;

    int ps = pre_sampled[b];
    int eos_new = (ps == 6) ? eos[b] : ps;
    int cl = c_len[b];
    int flag = (pre_seg[b] != 5) || (ps == 6) || (cl > 20);
    int idx = index[b] + (flag ? 1 : 0);
    if (idx > T_ - 1) idx = T_ - 1;
    cl = flag ? cl : cl + 1;
    int seg = templates[b * T_ + idx];
    int smp = flag ? seg : sampled[b];

    out_idxs[b * ML_ + t] = (float)smp;               // pre_sampled output column t
    pre_sampled[b] = smp; pre_seg[b] = seg; index[b] = idx;
    c_len[b] = cl; eos[b] = eos_new;
    keep[b] = ((smp == 5) || (smp == 6)) ? 1 : 0;
    int sin = (smp == 6) ? eos_new : smp;
    int e = (sin > V_) ? 3 : sin;
    if (e > V_ - 1) e = V_ - 1;                       // JAX gather clamp
    embidx[b] = e;
}

// ---------------------------------------------------------------------------
// Build x = [pre_hidden_sel, t_hidden, sel_read, emb]; updates pre_hidden
// ---------------------------------------------------------------------------
__global__ void k_build_x(const int* st, const float* __restrict__ temp_outputs,
                          const float* __restrict__ embedding,
                          float* __restrict__ hidden, float* __restrict__ pre_hidden,
                          const float* __restrict__ sel, float* __restrict__ x) {
    int tid = blockIdx.x * blockDim.x + threadIdx.x;
    if (tid >= B_ * D_) return;
    int b = tid / D_, j = tid - b * D_;
    const int* keep = st + 6*B_; const int* index = st + 4*B_; const int* embidx = st + 7*B_;
    float val;
    if (j < H_) {
        float ph = keep[b] ? pre_hidden[b * H_ + j] : hidden[b * H_ + j];
        pre_hidden[b * H_ + j] = ph;
        val = ph;
    } else if (j < 2 * H_) {
        val = temp_outputs[(b * T_ + index[b]) * H_ + (j - H_)];
    } else if (j < 3 * H_) {
        val = sel[b * H_ + (j - 2 * H_)];
    } else {
        val = embedding[embidx[b] * E_ + (j - 3 * H_)];
    }
    x[b * D_ + j] = val;
}

// ---------------------------------------------------------------------------
// GRU gates: hidden = (1-z)*n + z*pre_hidden ; also copies hidden into xcat hi
// ---------------------------------------------------------------------------
__global__ void k_gates(const float* __restrict__ gi, const float* __restrict__ gh,
                        const float* __restrict__ pre_hidden,
                        float* __restrict__ hidden, float* __restrict__ xcat) {
    int tid = blockIdx.x * blockDim.x + threadIdx.x;
    if (tid >= B_ * H_) return;
    int b = tid / H_, j = tid - b * H_;
    float ir = gi[b * G3_ + j],        hr = gh[b * G3_ + j];
    float iz = gi[b * G3_ + H_ + j],   hz = gh[b * G3_ + H_ + j];
    float in = gi[b * G3_ + 2*H_ + j], hn = gh[b * G3_ + 2*H_ + j];
    float r  = 1.0f / (1.0f + __expf(-(ir + hr)));
    float zg = 1.0f / (1.0f + __expf(-(iz + hz)));
    float n  = tanhf(in + r * hn);
    float h  = (1.0f - zg) * n + zg * pre_hidden[b * H_ + j];
    hidden[b * H_ + j] = h;
    xcat[b * 2*H_ + H_ + j] = h;
}

// ---------------------------------------------------------------------------
// Attention: scores->softmax->context (into xcat lo), raw copy scores css,
// plus sparse copy-score build (unique input token -> summed css).
// ---------------------------------------------------------------------------
__global__ void k_attn(const float* __restrict__ enc, const float* __restrict__ th,
                       const float* __restrict__ th2, const int* __restrict__ inputs,
                       float* __restrict__ xcat, float* __restrict__ css,
                       int* __restrict__ sp_idx, float* __restrict__ sp_val,
                       int* __restrict__ sp_cnt) {
    const int b = blockIdx.x, tid = threadIdx.x;
    __shared__ float sc[S_], sc2[S_], sth[H_], sth2[H_];
    __shared__ int sinp[S_];
    __shared__ int scnt;
    for (int h = tid; h < H_; h += blockDim.x) { sth[h] = th[b*H_+h]; sth2[h] = th2[b*H_+h]; }
    for (int s = tid; s < S_; s += blockDim.x) sinp[s] = inputs[b * S_ + s];
    if (tid == 0) scnt = 0;
    __syncthreads();
    for (int s = tid; s < S_; s += blockDim.x) {
        const float* e = enc + (size_t)(b * S_ + s) * H_;
        float d1 = 0.f, d2 = 0.f;
        for (int h = 0; h < H_; ++h) { d1 += e[h] * sth[h]; d2 += e[h] * sth2[h]; }
        sc[s] = d1; sc2[s] = d2;
    }
    __syncthreads();
    if (tid == 0) {
        float m = -3.0e38f;
        for (int s = 0; s < S_; ++s) m = fmaxf(m, sc[s]);
        float sum = 0.f;
        for (int s = 0; s < S_; ++s) { sc[s] = __expf(sc[s] - m); sum += sc[s]; }
        float inv = 1.0f / sum;
        for (int s = 0; s < S_; ++s) sc[s] *= inv;
    }
    // sparse copy-score: thread s owns token if first occurrence (and token != 0)
    if (tid < S_) {
        int v = sinp[tid];
        bool first = (v != 0);
        for (int s2 = 0; s2 < tid && first; ++s2) if (sinp[s2] == v) first = false;
        if (first) {
            float tot = 0.f;
            for (int s2 = 0; s2 < S_; ++s2) if (sinp[s2] == v) tot += sc2[s2];
            int slot = atomicAdd(&scnt, 1);
            sp_idx[b * S_ + slot] = v;
            sp_val[b * S_ + slot] = tot;
        }
    }
    __syncthreads();
    if (tid == 0) sp_cnt[b] = scnt;
    for (int h = tid; h < H_; h += blockDim.x) {
        float c = 0.f;
        for (int s = 0; s < S_; ++s) c += sc[s] * enc[(size_t)(b * S_ + s) * H_ + h];
        xcat[b * 2*H_ + h] = c;
    }
    for (int s = tid; s < S_; s += blockDim.x) css[b * S_ + s] = sc2[s];
}

// ---------------------------------------------------------------------------
// Softmax reductions: concat [gen | copy_scores] where copy_scores is NEG
// everywhere except the sparse entries (exp(NEG-m) underflows to 0 exactly).
// ---------------------------------------------------------------------------
__global__ void k_reduce(const float* __restrict__ gen, const float* __restrict__ sp_val,
                         const int* __restrict__ sp_cnt,
                         float* __restrict__ mx, float* __restrict__ denom,
                         ull* __restrict__ argkey) {
    const int b = blockIdx.x, tid = threadIdx.x;
    __shared__ float red[256];
    const int cnt = sp_cnt[b];
    float m = NEGV;                                  // NEG entries exist in concat
    for (int i = tid; i < V_;  i += 256) m = fmaxf(m, gen[(size_t)b * V_ + i]);
    for (int i = tid; i < cnt; i += 256) m = fmaxf(m, sp_val[b * S_ + i]);
    red[tid] = m; __syncthreads();
    for (int o = 128; o > 0; o >>= 1) { if (tid < o) red[tid] = fmaxf(red[tid], red[tid+o]); __syncthreads(); }
    m = red[0]; __syncthreads();
    float s = 0.f;
    for (int i = tid; i < V_;  i += 256) s += __expf(gen[(size_t)b * V_ + i] - m);
    for (int i = tid; i < cnt; i += 256) s += __expf(sp_val[b * S_ + i] - m);
    red[tid] = s; __syncthreads();
    for (int o = 128; o > 0; o >>= 1) { if (tid < o) red[tid] += red[tid+o]; __syncthreads(); }
    if (tid == 0) { mx[b] = m; denom[b] = red[0]; argkey[b] = 0ull; }
}

// Dense final write: final[i] = exp(gen[i]-m)/den (i<V), 0 otherwise.
// Sparse positions are fixed up (and re-argmaxed) in k_fix_sel.
__global__ void k_finalize(const float* __restrict__ gen,
                           const float* __restrict__ mx, const float* __restrict__ denom,
                           float* __restrict__ dout, ull* __restrict__ argkey, int t) {
    const int b = blockIdx.y, tid = threadIdx.x;
    const int i0 = blockIdx.x * 1024;
    const float m = mx[b], inv = 1.0f / denom[b];
    __shared__ ull kred[256];
    ull best = 0ull;
    for (int i = i0 + tid; i < i0 + 1024 && i < VS_; i += 256) {
        float p = (i < V_) ? __expf(gen[(size_t)b * V_ + i] - m) * inv : 0.0f;
        dout[(size_t)b * ML_ * VS_ + (size_t)t * VS_ + i] = p;
        ull key = ((ull)__float_as_uint(p) << 32) | (unsigned)(0x7FFFFFFFu - (unsigned)i);
        best = (key > best) ? key : best;
    }
    kred[tid] = best; __syncthreads();
    for (int o = 128; o > 0; o >>= 1) { if (tid < o && kred[tid+o] > kred[tid]) kred[tid] = kred[tid+o]; __syncthreads(); }
    if (tid == 0 && kred[0] != 0ull) atomicMax(&argkey[b], kred[0]);
}

// ---------------------------------------------------------------------------
// Sparse fix-up (+= copy prob at scattered tokens), final argmax, selective
// read: sel = normalize(pos*css) @ enc. One block owns one batch row.
// ---------------------------------------------------------------------------
__global__ void k_fix_sel(const float* __restrict__ gen, const int* __restrict__ sp_idx,
                          const float* __restrict__ sp_val, const int* __restrict__ sp_cnt,
                          const float* __restrict__ mx, const float* __restrict__ denom,
                          const ull* __restrict__ argkey, int* st,
                          const int* __restrict__ inputs, const float* __restrict__ css,
                          const float* __restrict__ enc, float* __restrict__ sel,
                          float* __restrict__ dout, int t) {
    const int b = blockIdx.x, tid = threadIdx.x;
    __shared__ ull kred[128];
    __shared__ float w[S_];
    __shared__ int smp;
    const int cnt = sp_cnt[b];
    const float m = mx[b], inv = 1.0f / denom[b];

    ull best = (tid == 0) ? argkey[b] : 0ull;        // dense winner from k_finalize
    if (tid < cnt) {
        int v = sp_idx[b * S_ + tid];
        float p = __expf(gen[(size_t)b * V_ + v] - m) * inv
                + __expf(sp_val[b * S_ + tid] - m) * inv;
        dout[(size_t)b * ML_ * VS_ + (size_t)t * VS_ + v] = p;   // corrected value
        ull key = ((ull)__float_as_uint(p) << 32) | (unsigned)(0x7FFFFFFFu - (unsigned)v);
        best = (key > best) ? key : best;
    }
    kred[tid] = best; __syncthreads();
    for (int o = 64; o > 0; o >>= 1) { if (tid < o && kred[tid+o] > kred[tid]) kred[tid] = kred[tid+o]; __syncthreads(); }

    if (tid == 0) {
        int i = (int)(0x7FFFFFFFu - (unsigned)(kred[0] & 0xFFFFFFFFull));
        smp = i;
        st[b] = i;                                   // sampled state for next k_ctrl
        float L = 0.f;
        for (int s = 0; s < S_; ++s) {
            float v = (inputs[b * S_ + s] == i) ? css[b * S_ + s] : 0.0f;
            w[s] = v; L += fabsf(v);
        }
        L = fmaxf(L, 1e-12f);
        for (int s = 0; s < S_; ++s) w[s] /= L;
    }
    __syncthreads();
    for (int h = tid; h < H_; h += blockDim.x) {
        float v = 0.f;
        for (int s = 0; s < S_; ++s) v += w[s] * enc[(size_t)(b * S_ + s) * H_ + h];
        sel[b * H_ + h] = v;
    }
}

// ---------------------------------------------------------------------------
// Host orchestration
// ---------------------------------------------------------------------------
extern "C" void kernel_launch(void* const* d_in, const int* in_sizes, int n_in,
                              void* d_out, int out_size, void* d_ws, size_t ws_size,
                              hipStream_t stream) {
    const float* enc   = (const float*)d_in[0];
    const float* tmo   = (const float*)d_in[1];
    const float* z     = (const float*)d_in[2];
    const float* emb   = (const float*)d_in[3];
    const float* w_ih  = (const float*)d_in[4];
    const float* w_hh  = (const float*)d_in[5];
    const float* b_ih  = (const float*)d_in[6];
    const float* b_hh  = (const float*)d_in[7];
    const float* attw  = (const float*)d_in[8];
    const float* attb  = (const float*)d_in[9];
    const float* cpyw  = (const float*)d_in[10];
    const float* cpyb  = (const float*)d_in[11];
    const float* outw  = (const float*)d_in[12];
    const float* outb  = (const float*)d_in[13];
    const int*   templ = (const int*)d_in[14];
    const int*   inp   = (const int*)d_in[15];

    float* dout     = (float*)d_out;                       // [B,ML,VS] then [B,ML]
    float* out_idxs = dout + (size_t)B_ * ML_ * VS_;

    // workspace carve-up (floats first, then 8B/4B-aligned tails)
    float* f = (float*)d_ws;
    float* hidden = f;            f += B_*H_;
    float* pre_h  = f;            f += B_*H_;
    float* sel    = f;            f += B_*H_;
    float* x      = f;            f += B_*D_;
    float* gi     = f;            f += B_*G3_;
    float* gh     = f;            f += B_*G3_;
    float* th     = f;            f += B_*H_;
    float* th2    = f;            f += B_*H_;
    float* xcat   = f;            f += B_*2*H_;
    float* css    = f;            f += B_*S_;
    float* sp_val = f;            f += B_*S_;
    float* gen    = f;            f += B_*V_;
    float* mx     = f;            f += B_;
    float* denom  = f;            f += B_;
    ull*   argkey = (ull*)f;                               // even float count -> 8B aligned
    int*   st     = (int*)(argkey + B_);                   // 8 arrays x 16 ints
    int*   sp_idx = st + 8*B_;
    int*   sp_cnt = sp_idx + B_*S_;

    // init (rewrites all persistent state + output row 0 every call)
    k_init_big<<<(B_*VS_ + 255)/256, 256, 0, stream>>>(dout, z, hidden, pre_h, sel);
    k_init_state<<<1, 32, 0, stream>>>(st, templ, out_idxs);

    for (int t = 1; t < ML_; ++t) {
        k_ctrl  <<<1, 32, 0, stream>>>(st, templ, out_idxs, t);
        k_build_x<<<(B_*D_ + 255)/256, 256, 0, stream>>>(st, tmo, emb, hidden, pre_h, sel, x);
        // gi = x @ W_ih^T + b_ih (96 tiles, K=1836)  ||  gh = pre_h @ W_hh^T + b_hh (96 tiles, K=512)
        k_gemm_wmma2<<<48, 128, 0, stream>>>(x, D_, w_ih, D_, gi, G3_, D_, 96, b_ih,
                                             pre_h, H_, w_hh, H_, gh, G3_, H_, 96, b_hh);
        k_gates<<<(B_*H_ + 255)/256, 256, 0, stream>>>(gi, gh, pre_h, hidden, xcat);
        // th = hidden @ attn_w^T + attn_b  ||  th2 = hidden @ copy_w^T + copy_b (32+32 tiles)
        k_gemm_wmma2<<<16, 128, 0, stream>>>(hidden, H_, attw, H_, th,  H_, H_, 32, attb,
                                             hidden, H_, cpyw, H_, th2, H_, H_, 32, cpyb);
        k_attn<<<B_, 128, 0, stream>>>(enc, th, th2, inp, xcat, css, sp_idx, sp_val, sp_cnt);
        // gen = [ctx|hidden] @ out_w^T + out_b (1875 tiles, K=1024); gen[:,0]=NEG
        k_gemm_wmma<<<469, 128, 0, stream>>>(xcat, 2*H_, outw, 2*H_, gen, V_, 2*H_, 1875, outb, 1);
        k_reduce<<<B_, 256, 0, stream>>>(gen, sp_val, sp_cnt, mx, denom, argkey);
        k_finalize<<<dim3(30, B_), 256, 0, stream>>>(gen, mx, denom, dout, argkey, t);
        k_fix_sel<<<B_, 128, 0, stream>>>(gen, sp_idx, sp_val, sp_cnt, mx, denom,
                                          argkey, st, inp, css, enc, sel, dout, t);
    }
}